// Quantizer_62534723829856
// MI455X (gfx1250) — compile-verified
//
#include <hip/hip_runtime.h>
#include <math.h>

// ---------------- problem constants ----------------
#define NROWS 16384           // B*H*W = 16*32*32
#define DIM   256             // latent dim
#define KCODES 8192           // codebook size
#define HW    1024            // 32*32
#define CHW   (DIM*HW)

typedef float v2f __attribute__((ext_vector_type(2)));
typedef float v8f __attribute__((ext_vector_type(8)));
typedef int   v4i __attribute__((ext_vector_type(4)));

// LDS row stride 260 floats -> bank = (4*row + col) % 64: conflict-free for the
// 16-row broadcast read pattern of WMMA A/B fragment loads; 260*4 = 1040 = 65*16,
// so rows stay 16B-aligned for b128 staging.
#define APAD 260

// --------- CDNA5 async global->LDS staging (guarded; falls back to plain copy) ----
#if defined(__gfx1250__) && __has_builtin(__builtin_amdgcn_global_load_async_to_lds_b128)
#define USE_ASYNC_LDS 1
#else
#define USE_ASYNC_LDS 0
#endif

#if USE_ASYNC_LDS
#if __has_builtin(__builtin_amdgcn_s_wait_asynccnt)
#define WAIT_ASYNC(n) __builtin_amdgcn_s_wait_asynccnt(n)
#else
#define WAIT_ASYNC(n) asm volatile("s_wait_asynccnt %0" ::"i"(n) : "memory")
#endif
#else
#define WAIT_ASYNC(n) ((void)0)
#endif

// ---------------- tiny utility kernels ----------------
__global__ void k_zero(float* __restrict__ p, int n) {
  int i = blockIdx.x * blockDim.x + threadIdx.x;
  if (i < n) p[i] = 0.0f;
}

// one wave per row: out[r] = sum_c M[r,c]^2   (used for ||e||^2 and ||x_rot||^2)
__global__ void k_rownorm(const float* __restrict__ M, float* __restrict__ out, int nrows) {
  int wv   = (blockIdx.x * blockDim.x + threadIdx.x) >> 5;
  int lane = threadIdx.x & 31;
  if (wv >= nrows) return;
  const float* r = M + (size_t)wv * DIM;
  float s = 0.0f;
  for (int c = lane; c < DIM; c += 32) { float v = r[c]; s += v * v; }
  #pragma unroll
  for (int o = 16; o > 0; o >>= 1) s += __shfl_xor(s, o, 32);
  if (lane == 0) out[wv] = s;
}

// ---------------- rotation GEMM: x_rot = x_flat @ R  (fp32 WMMA 16x16x4) ----------------
// x is (B,C,H,W); x_flat[n,c] = x[b*CHW + c*HW + p], n = b*1024 + p.
// grid = (NROWS/16, 2), block = 256 (8 waves, each wave one 16x16 column tile).
__global__ void __launch_bounds__(256) k_rotate(const float* __restrict__ x,
                                                const float* __restrict__ R,
                                                float* __restrict__ xrot) {
  __shared__ float As[16 * APAD];
  const int tid = threadIdx.x;
  const int rowbase = blockIdx.x * 16;
  const int b  = rowbase >> 10;
  const int p0 = rowbase & 1023;

  // stage the 16x256 input strip (transposed gather from NCHW) into LDS
  {
    int m  = tid & 15;      // consecutive lanes -> consecutive global addresses
    int c0 = tid >> 4;
    #pragma unroll
    for (int pass = 0; pass < 16; ++pass) {
      int c = c0 + pass * 16;
      As[m * APAD + c] = x[b * CHW + c * HW + p0 + m];
    }
  }
  __syncthreads();

  const int lane = tid & 31, half = lane >> 4, l15 = lane & 15;
  const int wv = tid >> 5;
  const int colbase = (blockIdx.y * 8 + wv) * 16;

  v8f acc = {};
  for (int kb = 0; kb < DIM; kb += 4) {
    const int kk = kb + 2 * half;
    v2f a, bv;
    a.x  = As[l15 * APAD + kk];
    a.y  = As[l15 * APAD + kk + 1];
    bv.x = R[kk * DIM + colbase + l15];
    bv.y = R[(kk + 1) * DIM + colbase + l15];
    acc = __builtin_amdgcn_wmma_f32_16x16x4_f32(false, a, false, bv, (short)0, acc,
                                                false, false);
  }
  #pragma unroll
  for (int v = 0; v < 8; ++v) {
    int m = v + 8 * half;
    xrot[(size_t)(rowbase + m) * DIM + colbase + l15] = acc[v];
  }
}

// ---------------- fused distance GEMM + argmin (fp32 WMMA 16x16x4) ----------------
// Block = 256 threads (8 waves) covering 64 rows (4 row tiles); all waves march over
// the SAME 32-code E tile per iteration (waves = 4 row tiles x 2 code halves), so the
// E tile is staged once per block into double-buffered LDS via async global->LDS
// copies (ASYNCcnt), overlapping tile i+1's copy with tile i's 64 WMMAs.
#define DROWS 64
#define ETILE 32

// stage 32 codebook rows (32x256 f32 = 32KB) into padded LDS; 256 threads,
// 8 threads/row, 8 x b128 chunks per thread, fully coalesced 128B bursts.
__device__ __forceinline__ void stage_e_tile(const float* __restrict__ E,
                                             float* __restrict__ sE,
                                             int cbase, int tid) {
  const int r  = tid >> 3;          // 0..31
  const int c0 = (tid & 7) * 4;     // float offset in row, +32 per chunk
  const float* gp = E + (size_t)(cbase + r) * DIM + c0;
  float* lp = sE + r * APAD + c0;
#if USE_ASYNC_LDS
  #pragma unroll
  for (int i = 0; i < 8; ++i)
    __builtin_amdgcn_global_load_async_to_lds_b128(
        (__attribute__((address_space(1))) v4i*)(gp + i * 32),
        (__attribute__((address_space(3))) v4i*)(lp + i * 32), 0, 0);
#else
  #pragma unroll
  for (int i = 0; i < 8; ++i)
    *(float4*)(lp + i * 32) = *(const float4*)(gp + i * 32);
#endif
}

__global__ void __launch_bounds__(256) k_dist_argmin(const float* __restrict__ xrot,
                                                     const float* __restrict__ E,
                                                     const float* __restrict__ xnorm,
                                                     const float* __restrict__ enorm,
                                                     int* __restrict__ enc_i,
                                                     float* __restrict__ enc_f) {
  __shared__ float As[DROWS * APAD];          // 64x256 x_rot strip      (66.5 KB)
  __shared__ float Eb[2 * ETILE * APAD];      // double-buffered E tile  (66.5 KB)
  __shared__ float sxn[DROWS];
  __shared__ float sD[8][16];
  __shared__ int   sI[8][16];

  const int tid = threadIdx.x;
  const int rowbase = blockIdx.x * DROWS;

  // stage A (64x256, row-major contiguous) with b128 chunks
  for (int i = tid; i < DROWS * (DIM / 4); i += 256) {
    int m = i >> 6, ch = i & 63;
    *(float4*)(As + m * APAD + ch * 4) =
        *(const float4*)(xrot + (size_t)(rowbase + m) * DIM + ch * 4);
  }
  if (tid < DROWS) sxn[tid] = xnorm[rowbase + tid];

  stage_e_tile(E, Eb, 0, tid);                // prologue: buffer 0, tile 0

  const int lane = tid & 31, half = lane >> 4, l15 = lane & 15;
  const int wv = tid >> 5;
  const int rt  = wv & 3;                     // row tile 0..3
  const int ch8 = wv >> 2;                    // code half 0..1
  const int arow = (rt * 16 + l15) * APAD;    // A fragment row base in LDS
  const int codeloc = ch8 * 16 + l15;         // lane's row in the E tile

  float bestD[8]; int bestI[8];
  #pragma unroll
  for (int v = 0; v < 8; ++v) { bestD[v] = 3.4e38f; bestI[v] = 0x7fffffff; }

  const int NIT = KCODES / ETILE;             // 256 iterations
  for (int it = 0; it < NIT; ++it) {
    if (it + 1 < NIT) {
      stage_e_tile(E, Eb + ((it + 1) & 1) * (ETILE * APAD), (it + 1) * ETILE, tid);
      WAIT_ASYNC(8);                          // first batch (current tile) complete
    } else {
      WAIT_ASYNC(0);
    }
    __syncthreads();                          // all threads' copies visible

    const float* Ecur = Eb + (it & 1) * (ETILE * APAD);
    const int code = it * ETILE + codeloc;
    const float en = enorm[code];

    v8f acc = {};
    #pragma unroll 8
    for (int kb = 0; kb < DIM; kb += 4) {
      const int kk = kb + 2 * half;
      v2f a, bv;
      a.x  = As[arow + kk];
      a.y  = As[arow + kk + 1];
      bv.x = Ecur[codeloc * APAD + kk];
      bv.y = Ecur[codeloc * APAD + kk + 1];
      acc = __builtin_amdgcn_wmma_f32_16x16x4_f32(false, a, false, bv, (short)0, acc,
                                                  false, false);
    }
    #pragma unroll
    for (int v = 0; v < 8; ++v) {
      int m = v + 8 * half;
      float d = sxn[rt * 16 + m] + en - 2.0f * acc[v];
      if (d < bestD[v] || (d == bestD[v] && code < bestI[v])) {
        bestD[v] = d; bestI[v] = code;
      }
    }
    __syncthreads();                          // nobody overwrites tile being read
  }

  // reduce across the 16 column lanes of each half-wave (wave32: xor stays in half)
  #pragma unroll
  for (int off = 1; off < 16; off <<= 1) {
    #pragma unroll
    for (int v = 0; v < 8; ++v) {
      float d2 = __shfl_xor(bestD[v], off, 32);
      int   i2 = __shfl_xor(bestI[v], off, 32);
      if (d2 < bestD[v] || (d2 == bestD[v] && i2 < bestI[v])) {
        bestD[v] = d2; bestI[v] = i2;
      }
    }
  }
  if (l15 == 0) {
    #pragma unroll
    for (int v = 0; v < 8; ++v) {
      int m = v + 8 * half;
      sD[wv][m] = bestD[v]; sI[wv][m] = bestI[v];
    }
  }
  __syncthreads();
  if (tid < DROWS) {                          // merge waves {rt, rt+4} per row
    int rt2 = tid >> 4, m = tid & 15;
    float bd = sD[rt2][m];     int bi = sI[rt2][m];
    float d2 = sD[rt2 + 4][m]; int i2 = sI[rt2 + 4][m];
    if (d2 < bd || (d2 == bd && i2 < bi)) { bd = d2; bi = i2; }
    enc_i[rowbase + tid] = bi;
    enc_f[rowbase + tid] = (float)bi;
  }
}

// ---------------- gather + MSE + scatter(counts, dw) ----------------
// grid = NROWS blocks of 256: block n handles all 256 channels of token n.
__global__ void __launch_bounds__(256) k_gather_loss(const float* __restrict__ x,
                                                     const float* __restrict__ E,
                                                     const float* __restrict__ xrot,
                                                     const int* __restrict__ enc_i,
                                                     float* __restrict__ qout,
                                                     float* __restrict__ dw,
                                                     float* __restrict__ counts,
                                                     float* __restrict__ acc) {
  __shared__ float red[256];
  const int tid = threadIdx.x;
  const int n = blockIdx.x;
  const int c = tid;
  const int idx = enc_i[n];
  const int b = n >> 10, p = n & 1023;

  float q  = E[(size_t)idx * DIM + c];
  float xv = x[b * CHW + c * HW + p];
  qout[b * CHW + c * HW + p] = q;          // quant_out_st forward value == quant
  float diff = q - xv;
  red[tid] = diff * diff;

  atomicAdd(&dw[(size_t)idx * DIM + c], xrot[(size_t)n * DIM + c]);
  if (tid == 0) atomicAdd(&counts[idx], 1.0f);

  __syncthreads();
  for (int s = 128; s > 0; s >>= 1) {
    if (tid < s) red[tid] += red[tid + s];
    __syncthreads();
  }
  if (tid == 0) atomicAdd(&acc[0], red[0]);
}

// ---------------- entropy + pre-normalized new_cluster + n_tot ----------------
__global__ void __launch_bounds__(256) k_entropy_nc(const float* __restrict__ counts,
                                                    const float* __restrict__ ema_cs,
                                                    float* __restrict__ nc_out,
                                                    float* __restrict__ acc) {
  __shared__ float r1[256], r2[256];
  const int tid = threadIdx.x;
  const int k = blockIdx.x * 256 + tid;
  float cnt = counts[k];
  float p = cnt * (1.0f / (float)NROWS);
  float ent = p * logf(p + 1e-10f);
  float nc = ema_cs[k] * 0.99f + 0.01f * cnt;
  nc_out[k] = nc;
  r1[tid] = ent; r2[tid] = nc;
  __syncthreads();
  for (int s = 128; s > 0; s >>= 1) {
    if (tid < s) { r1[tid] += r1[tid + s]; r2[tid] += r2[tid + s]; }
    __syncthreads();
  }
  if (tid == 0) { atomicAdd(&acc[1], r1[0]); atomicAdd(&acc[2], r2[0]); }
}

__global__ void k_scalars(const float* __restrict__ acc, float* __restrict__ o) {
  if (threadIdx.x == 0 && blockIdx.x == 0) {
    float mse = acc[0] * (1.0f / (float)(NROWS * (long)DIM));
    o[0] = mse;                // codebook_loss
    o[1] = 0.25f * mse;        // commitment_loss (BETA)
    float el = -acc[1];        // * DIVERSITY_WEIGHT (1.0)
    o[2] = el;                 // entropy_loss
    o[3] = expf(el);           // perplexity
  }
}

__global__ void k_ncnorm(float* __restrict__ nc, const float* __restrict__ acc) {
  int k = blockIdx.x * 256 + threadIdx.x;
  if (k >= KCODES) return;
  float ntot = acc[2];
  nc[k] = (nc[k] + 1e-5f) / (ntot + (float)KCODES * 1e-5f) * ntot;
}

__global__ void __launch_bounds__(256) k_ema(const float* __restrict__ ema_w,
                                             const float* __restrict__ dw,
                                             const float* __restrict__ nc,
                                             float* __restrict__ new_ema_w,
                                             float* __restrict__ new_emb) {
  int gid = blockIdx.x * 256 + threadIdx.x;   // KCODES*DIM
  int k = gid >> 8;
  float ew = ema_w[gid] * 0.99f + 0.01f * dw[gid];
  new_ema_w[gid] = ew;
  new_emb[gid] = ew / nc[k];
}

// ---------------- host-side orchestration ----------------
extern "C" void kernel_launch(void* const* d_in, const int* in_sizes, int n_in,
                              void* d_out, int out_size, void* d_ws, size_t ws_size,
                              hipStream_t stream) {
  (void)in_sizes; (void)n_in; (void)out_size; (void)ws_size;
  const float* x      = (const float*)d_in[0];   // (16,256,32,32)
  const float* E      = (const float*)d_in[1];   // (8192,256)
  const float* R      = (const float*)d_in[2];   // (256,256)
  const float* ema_cs = (const float*)d_in[3];   // (8192,)
  const float* ema_w  = (const float*)d_in[4];   // (8192,256)
  float* out = (float*)d_out;
  float* w   = (float*)d_ws;

  // workspace layout (floats)
  const size_t XROT   = 0;                               // 4,194,304
  const size_t DW     = XROT + (size_t)NROWS * DIM;      // 2,097,152
  const size_t COUNTS = DW + (size_t)KCODES * DIM;       // 8,192
  const size_t ACC    = COUNTS + KCODES;                 // 16
  const size_t ENORM  = ACC + 16;                        // 8,192
  const size_t XNORM  = ENORM + KCODES;                  // 16,384
  const size_t ENCI   = XNORM + NROWS;                   // 16,384 (int)

  // output layout (floats, concatenated in reference return order)
  const size_t QOUT  = 0;
  const size_t CB    = QOUT + (size_t)NROWS * DIM;       // 4 scalars start here
  const size_t ENC   = CB + 4;
  const size_t NEMB  = ENC + NROWS;
  const size_t NEMAW = NEMB + (size_t)KCODES * DIM;
  const size_t NCL   = NEMAW + (size_t)KCODES * DIM;

  // zero the accumulation regions (dw + counts + acc are contiguous)
  {
    int nz = (int)(KCODES * DIM + KCODES + 16);
    k_zero<<<(nz + 255) / 256, 256, 0, stream>>>(w + DW, nz);
  }
  k_rownorm<<<(KCODES * 32) / 256, 256, 0, stream>>>(E, w + ENORM, KCODES);
  k_rotate<<<dim3(NROWS / 16, 2), 256, 0, stream>>>(x, R, w + XROT);
  k_rownorm<<<(NROWS * 32) / 256, 256, 0, stream>>>(w + XROT, w + XNORM, NROWS);
  k_dist_argmin<<<NROWS / DROWS, 256, 0, stream>>>(w + XROT, E, w + XNORM, w + ENORM,
                                                   (int*)(w + ENCI), out + ENC);
  k_gather_loss<<<NROWS, 256, 0, stream>>>(x, E, w + XROT, (const int*)(w + ENCI),
                                           out + QOUT, w + DW, w + COUNTS, w + ACC);
  k_entropy_nc<<<KCODES / 256, 256, 0, stream>>>(w + COUNTS, ema_cs, out + NCL, w + ACC);
  k_scalars<<<1, 32, 0, stream>>>(w + ACC, out + CB);
  k_ncnorm<<<32, 256, 0, stream>>>(out + NCL, w + ACC);
  k_ema<<<(KCODES * DIM) / 256, 256, 0, stream>>>(ema_w, w + DW, out + NCL,
                                                  out + NEMAW, out + NEMB);
}